// NNUE_3152505995829
// MI455X (gfx1250) — compile-verified
//
#include <hip/hip_runtime.h>

// NNUE forward: B=65536 positions, 2 bags x 32 indices, acc=128, buckets=4.
// Strategy: one wave32 per position; gather acc_w rows as global_load_b128
// (SGPR row base via v_readlane + per-lane 16B offset); psqt via one b128
// per lane + shfl_xor reduce; final 256->4 layer via V_WMMA_F32_16X16X4_F32
// on 16 positions per block (full f32, matches reference numerics).

#define NFEAT 20480
#define NACC  128
#define NBUCK 4
#define BAG   32
#define POS_PER_BLOCK 16
#define THREADS (POS_PER_BLOCK * 32)

typedef __attribute__((ext_vector_type(2))) float v2f;
typedef __attribute__((ext_vector_type(8))) float v8f;

#if defined(__has_builtin)
#  if __has_builtin(__builtin_amdgcn_wmma_f32_16x16x4_f32)
#    define HAVE_WMMA_F32X4 1
#  endif
#endif

__device__ __forceinline__ float4 ld4(const float* p) {
  return *(const float4*)p;
}

__global__ __launch_bounds__(THREADS) void nnue_fwd(
    const int* __restrict__ w_cols, const int* __restrict__ b_cols,
    const float* __restrict__ stms, const float* __restrict__ psqt_w,
    const float* __restrict__ acc_w, const float* __restrict__ acc_bias,
    const float* __restrict__ layer_w, const float* __restrict__ layer_b,
    float* __restrict__ out, int n_pos)
{
  // Row stride 260 floats (256 + 4 pad): stride % 64 == 4 -> the WMMA A-frag
  // column reads (16 rows, same k) land on distinct LDS banks.
  __shared__ float sBlend[POS_PER_BLOCK][260];
  __shared__ float sPsqt[POS_PER_BLOCK][NBUCK];
  __shared__ float sSign[POS_PER_BLOCK];

  const int wave = threadIdx.x >> 5;
  const int lane = threadIdx.x & 31;
  const int pos  = blockIdx.x * POS_PER_BLOCK + wave;

  if (pos < n_pos) {
    const size_t base = (size_t)pos * BAG;
    // One index per lane (also feeds psqt gather), then readlane-broadcast.
    const int wcolL = w_cols[base + lane];
    const int bcolL = b_cols[base + lane];

    // psqt partial: lane j handles bag element j (4 buckets per lane).
    float4 p;
    {
      float4 pw = ld4(psqt_w + (size_t)wcolL * NBUCK);
      float4 pb = ld4(psqt_w + (size_t)bcolL * NBUCK);
      p.x = pw.x - pb.x; p.y = pw.y - pb.y;
      p.z = pw.z - pb.z; p.w = pw.w - pb.w;
    }

    // Accumulator gather-sums: 4 floats per lane, b128 per row per lane.
    float4 wa = make_float4(0.f, 0.f, 0.f, 0.f);
    float4 ba = make_float4(0.f, 0.f, 0.f, 0.f);
    const int laneOff = lane * 4;
#pragma unroll
    for (int j = 0; j < BAG; ++j) {
      const int wc = __builtin_amdgcn_readlane(wcolL, j);  // SGPR row base
      float4 r = ld4(acc_w + (size_t)wc * NACC + laneOff);
      wa.x += r.x; wa.y += r.y; wa.z += r.z; wa.w += r.w;
    }
#pragma unroll
    for (int j = 0; j < BAG; ++j) {
      const int bc = __builtin_amdgcn_readlane(bcolL, j);
      float4 r = ld4(acc_w + (size_t)bc * NACC + laneOff);
      ba.x += r.x; ba.y += r.y; ba.z += r.z; ba.w += r.w;
    }

    // bias + clamp [0,1]
    float4 bias = ld4(acc_bias + laneOff);
    wa.x = fminf(fmaxf(wa.x + bias.x, 0.f), 1.f);
    wa.y = fminf(fmaxf(wa.y + bias.y, 0.f), 1.f);
    wa.z = fminf(fmaxf(wa.z + bias.z, 0.f), 1.f);
    wa.w = fminf(fmaxf(wa.w + bias.w, 0.f), 1.f);
    ba.x = fminf(fmaxf(ba.x + bias.x, 0.f), 1.f);
    ba.y = fminf(fmaxf(ba.y + bias.y, 0.f), 1.f);
    ba.z = fminf(fmaxf(ba.z + bias.z, 0.f), 1.f);
    ba.w = fminf(fmaxf(ba.w + bias.w, 0.f), 1.f);

    // stm blend: s in {0,1} -> pure select + sign flip.
    const float s = stms[pos];
    const bool flip = (s != 0.0f);
    float4 first  = flip ? ba : wa;
    float4 second = flip ? wa : ba;
    float* row = &sBlend[wave][0];
    *(float4*)(row + 4 * lane)        = first;   // blended[0..127]
    *(float4*)(row + NACC + 4 * lane) = second;  // blended[128..255]

    // reduce psqt partials across 32 lanes
#pragma unroll
    for (int off = 16; off > 0; off >>= 1) {
      p.x += __shfl_xor(p.x, off, 32);
      p.y += __shfl_xor(p.y, off, 32);
      p.z += __shfl_xor(p.z, off, 32);
      p.w += __shfl_xor(p.w, off, 32);
    }
    if (lane == 0) {
      sPsqt[wave][0] = p.x; sPsqt[wave][1] = p.y;
      sPsqt[wave][2] = p.z; sPsqt[wave][3] = p.w;
      sSign[wave] = flip ? -1.0f : 1.0f;
    }
  }
  __syncthreads();

  // Final layer: out[16,4] = blended[16,256] @ layer_w^T[256,4] (N padded to 16)
  if (wave == 0) {
    const int n  = lane & 15;        // output column (bucket, valid < 4)
    const int hi = lane >> 4;        // half selector per WMMA f32 A/B layout
    const int nq = (n < NBUCK) ? n : 0;
    const float nmask = (n < NBUCK) ? 1.0f : 0.0f;
    const float* lw = layer_w + (size_t)nq * (2 * NACC);
    const float* arow = &sBlend[n][0];   // A-matrix row m == lane&15

#ifdef HAVE_WMMA_F32X4
    v8f c = {0.f, 0.f, 0.f, 0.f, 0.f, 0.f, 0.f, 0.f};
#pragma unroll 8
    for (int k4 = 0; k4 < (2 * NACC) / 4; ++k4) {
      const int kb = k4 * 4 + hi * 2;  // lanes 0-15: K=0,1 ; lanes 16-31: K=2,3
      v2f a; a.x = arow[kb];        a.y = arow[kb + 1];
      v2f b; b.x = lw[kb] * nmask;  b.y = lw[kb + 1] * nmask;
      c = __builtin_amdgcn_wmma_f32_16x16x4_f32(
              false, a, false, b, (short)0, c, false, false);
    }
    if (n < NBUCK) {
      const float lb = layer_b[n];
#pragma unroll
      for (int r = 0; r < 8; ++r) {
        const int m = r + hi * 8;    // D row per C/D 16x16 f32 layout
        const int opos = blockIdx.x * POS_PER_BLOCK + m;
        if (opos < n_pos)
          out[(size_t)opos * NBUCK + n] =
              sPsqt[m][n] + sSign[m] * (c[r] + lb);
      }
    }
#else
    // VALU fallback (also proves correct data layout if WMMA missing)
    if (hi == 0) {
      const int m = n;
      const int opos = blockIdx.x * POS_PER_BLOCK + m;
      if (opos < n_pos) {
        const float* brow = &sBlend[m][0];
        const float sign = sSign[m];
#pragma unroll
        for (int bkt = 0; bkt < NBUCK; ++bkt) {
          float acc = layer_b[bkt];
          for (int k = 0; k < 2 * NACC; ++k)
            acc += brow[k] * layer_w[bkt * (2 * NACC) + k];
          out[(size_t)opos * NBUCK + bkt] = sPsqt[m][bkt] + sign * acc;
        }
      }
    }
#endif
  }
}

extern "C" void kernel_launch(void* const* d_in, const int* in_sizes, int n_in,
                              void* d_out, int out_size, void* d_ws, size_t ws_size,
                              hipStream_t stream) {
  // setup_inputs order:
  // 0 w_offset (unused), 1 w_cols, 2 b_offset (unused), 3 b_cols,
  // 4 stms, 5 psqt_w, 6 acc_w, 7 acc_bias, 8 layer_w, 9 layer_b
  const int*   w_cols   = (const int*)d_in[1];
  const int*   b_cols   = (const int*)d_in[3];
  const float* stms     = (const float*)d_in[4];
  const float* psqt_w   = (const float*)d_in[5];
  const float* acc_w    = (const float*)d_in[6];
  const float* acc_bias = (const float*)d_in[7];
  const float* layer_w  = (const float*)d_in[8];
  const float* layer_b  = (const float*)d_in[9];
  float* out = (float*)d_out;

  const int n_pos = in_sizes[4];  // B (stms has one entry per position)
  const int blocks = (n_pos + POS_PER_BLOCK - 1) / POS_PER_BLOCK;
  nnue_fwd<<<blocks, THREADS, 0, stream>>>(
      w_cols, b_cols, stms, psqt_w, acc_w, acc_bias, layer_w, layer_b,
      out, n_pos);
}